// EigenAttention_40862318854763
// MI455X (gfx1250) — compile-verified
//
#include <hip/hip_runtime.h>
#include <hip/hip_bf16.h>
#include <math.h>

// ---------------------------------------------------------------------------
// EigenAttention for MI455X (gfx1250, wave32, WMMA f32 16x16x4)
//   B=4, SQ=SK=4096, HIDDEN=1024, K=64
// Memory-bound on the 268MB attn matrix that must be returned; all GEMMs use
// V_WMMA_F32_16X16X4_F32 to keep full fp32 accuracy (compute is not the
// bottleneck: ~26 GFLOP vs ~1.3 GB of traffic @ 23.3 TB/s).
// ---------------------------------------------------------------------------

typedef __attribute__((ext_vector_type(2))) float v2f;
typedef __attribute__((ext_vector_type(8))) float v8f;

#define HIDDEN 1024
#define KDIM   64
#define BATCH  4
#define SEQ    4096
#define NROWS  (BATCH * SEQ)   // 16384 flat (b,s) rows

__device__ __forceinline__ v8f wmma4(v2f a, v2f b, v8f c) {
  // D = A(16x4 f32) x B(4x16 f32) + C(16x16 f32)
  return __builtin_amdgcn_wmma_f32_16x16x4_f32(
      /*neg_a=*/false, a, /*neg_b=*/false, b,
      /*c_mod=*/(short)0, c, /*reuse_a=*/false, /*reuse_b=*/false);
}

// ---------------------------------------------------------------------------
// Kernel 1: projections  q_c = (q - qm) @ Wq^T  (and k_c, v_c)
// grid.x = NROWS/128, grid.y = 3 (q,k,v). block = 256 (8 waves).
// Each wave: 16 rows x 64 cols, K-reduction 1024 -> 256 WMMA steps x 4 tiles.
// v_c is stored TRANSPOSED per batch: vT[b][c][s] so the AV GEMM gets
// contiguous B-fragment loads.
// ---------------------------------------------------------------------------
__global__ void eig_proj_kernel(const float* __restrict__ q,
                                const float* __restrict__ k,
                                const float* __restrict__ v,
                                const float* __restrict__ Wq,
                                const float* __restrict__ Wk,
                                const float* __restrict__ Wv,
                                const float* __restrict__ qm,
                                const float* __restrict__ km,
                                const float* __restrict__ vm,
                                float* __restrict__ q_c,
                                float* __restrict__ k_c,
                                float* __restrict__ vT) {
  const int z = blockIdx.y;
  const float* __restrict__ X  = (z == 0) ? q  : (z == 1) ? k  : v;
  const float* __restrict__ W  = (z == 0) ? Wq : (z == 1) ? Wk : Wv;
  const float* __restrict__ Mn = (z == 0) ? qm : (z == 1) ? km : vm;

  const int lane = threadIdx.x & 31;
  const int wv   = threadIdx.x >> 5;       // wave 0..7
  const int l16  = lane & 15;
  const int lhi  = lane >> 4;              // 0 or 1
  const int m    = blockIdx.x * 128 + wv * 16 + l16;   // A-matrix row

  v8f C[4] = {};

#pragma unroll 4
  for (int kk = 0; kk < HIDDEN / 4; ++kk) {
    const int kd = kk * 4 + lhi * 2;
    v2f a = *(const v2f*)(X + (size_t)m * HIDDEN + kd);
    a.x -= Mn[kd];
    a.y -= Mn[kd + 1];
#pragma unroll
    for (int ct = 0; ct < 4; ++ct) {
      const int n = ct * 16 + l16;                      // output col (eigen dim)
      v2f bf = *(const v2f*)(W + (size_t)n * HIDDEN + kd);
      C[ct] = wmma4(a, bf, C[ct]);
    }
  }

  if (z < 2) {
    float* __restrict__ O = (z == 0) ? q_c : k_c;       // row-major (NROWS,64)
#pragma unroll
    for (int ct = 0; ct < 4; ++ct)
#pragma unroll
      for (int r = 0; r < 8; ++r) {
        const int row = blockIdx.x * 128 + wv * 16 + r + lhi * 8;
        O[(size_t)row * KDIM + ct * 16 + l16] = C[ct][r];
      }
  } else {
    // transposed store: vT[(b*64 + c)*4096 + s]
#pragma unroll
    for (int ct = 0; ct < 4; ++ct)
#pragma unroll
      for (int r = 0; r < 8; ++r) {
        const int row = blockIdx.x * 128 + wv * 16 + r + lhi * 8;
        const int b = row >> 12;           // row / 4096
        const int s = row & 4095;
        vT[((size_t)b * KDIM + ct * 16 + l16) * SEQ + s] = C[ct][r];
      }
  }
}

// ---------------------------------------------------------------------------
// Kernel 2: raw scores = (q_c @ k_c^T) / sqrt(K), written into attn region.
// grid = (SK/128, SQ/128, B), block = 256. Wave: 16 q-rows x 128 k-cols.
// q_c/k_c are 8MB total -> L2 resident; fragments loaded directly (float2).
// ---------------------------------------------------------------------------
__global__ void eig_scores_kernel(const float* __restrict__ q_c,
                                  const float* __restrict__ k_c,
                                  float* __restrict__ attn) {
  const int b    = blockIdx.z;
  const int lane = threadIdx.x & 31;
  const int wv   = threadIdx.x >> 5;
  const int l16  = lane & 15;
  const int lhi  = lane >> 4;

  const float* __restrict__ qb = q_c + (size_t)b * SEQ * KDIM;
  const float* __restrict__ kb = k_c + (size_t)b * SEQ * KDIM;
  float* __restrict__ ab = attn + (size_t)b * SEQ * SEQ;

  const int q0 = blockIdx.y * 128 + wv * 16;
  const int c0 = blockIdx.x * 128;

  v8f C[8] = {};

#pragma unroll
  for (int kk = 0; kk < KDIM / 4; ++kk) {               // 16 steps
    const int kd = kk * 4 + lhi * 2;
    v2f a = *(const v2f*)(qb + (size_t)(q0 + l16) * KDIM + kd);
#pragma unroll
    for (int ct = 0; ct < 8; ++ct) {
      v2f bf = *(const v2f*)(kb + (size_t)(c0 + ct * 16 + l16) * KDIM + kd);
      C[ct] = wmma4(a, bf, C[ct]);
    }
  }

  const float scale = 0.125f;                           // 1/sqrt(64)
#pragma unroll
  for (int ct = 0; ct < 8; ++ct)
#pragma unroll
    for (int r = 0; r < 8; ++r) {
      const int qr = q0 + r + lhi * 8;
      const int cc = c0 + ct * 16 + l16;
      ab[(size_t)qr * SEQ + cc] = C[ct][r] * scale;
    }
}

// ---------------------------------------------------------------------------
// Kernel 3: in-place row softmax over 4096 elements. One block per row.
// ---------------------------------------------------------------------------
__global__ void eig_softmax_kernel(float* __restrict__ attn) {
  __shared__ float red[256];
  const int tid = threadIdx.x;
  float* __restrict__ p = attn + (size_t)blockIdx.x * SEQ;

  float vals[16];
  float mx = -INFINITY;
#pragma unroll
  for (int i = 0; i < 16; ++i) {
    vals[i] = p[tid + i * 256];
    mx = fmaxf(mx, vals[i]);
  }
  red[tid] = mx;
  __syncthreads();
  for (int s = 128; s > 0; s >>= 1) {
    if (tid < s) red[tid] = fmaxf(red[tid], red[tid + s]);
    __syncthreads();
  }
  mx = red[0];
  __syncthreads();

  float sum = 0.f;
#pragma unroll
  for (int i = 0; i < 16; ++i) {
    vals[i] = __expf(vals[i] - mx);
    sum += vals[i];
  }
  red[tid] = sum;
  __syncthreads();
  for (int s = 128; s > 0; s >>= 1) {
    if (tid < s) red[tid] += red[tid + s];
    __syncthreads();
  }
  const float inv = 1.0f / red[0];

#pragma unroll
  for (int i = 0; i < 16; ++i) p[tid + i * 256] = vals[i] * inv;
}

// ---------------------------------------------------------------------------
// Kernel 4: out_c = attn @ v_c  (M=4096/b, N=64, K=4096).
// grid = (SQ/128, B), block = 256. Wave: 16 rows x 64 cols, 1024 WMMA steps.
// attn streamed once (268MB read); vT is 4MB -> L2 resident.
// ---------------------------------------------------------------------------
__global__ void eig_outc_kernel(const float* __restrict__ attn,
                                const float* __restrict__ vT,
                                float* __restrict__ outc) {
  const int b    = blockIdx.y;
  const int lane = threadIdx.x & 31;
  const int wv   = threadIdx.x >> 5;
  const int l16  = lane & 15;
  const int lhi  = lane >> 4;

  const float* __restrict__ ab = attn + (size_t)b * SEQ * SEQ;
  const float* __restrict__ vb = vT + (size_t)b * KDIM * SEQ;
  const int q0 = blockIdx.x * 128 + wv * 16;

  v8f C[4] = {};

#pragma unroll 4
  for (int kk = 0; kk < SEQ / 4; ++kk) {                // 1024 steps
    const int sd = kk * 4 + lhi * 2;
    v2f a = *(const v2f*)(ab + (size_t)(q0 + l16) * SEQ + sd);
#pragma unroll
    for (int ct = 0; ct < 4; ++ct) {
      v2f bf = *(const v2f*)(vb + (size_t)(ct * 16 + l16) * SEQ + sd);
      C[ct] = wmma4(a, bf, C[ct]);
    }
  }

#pragma unroll
  for (int ct = 0; ct < 4; ++ct)
#pragma unroll
    for (int r = 0; r < 8; ++r) {
      const int row = b * SEQ + q0 + r + lhi * 8;       // flat (b,s)
      outc[(size_t)row * KDIM + ct * 16 + l16] = C[ct][r];
    }
}

// ---------------------------------------------------------------------------
// Kernel 5: output = out_c @ W_out^T + b_out  (M=16384, N=1024, K=64).
// grid = NROWS/128, block = 256. A-fragments hoisted (16 x v2f), reused
// across 64 column tiles of 16.
// ---------------------------------------------------------------------------
__global__ void eig_outproj_kernel(const float* __restrict__ outc,
                                   const float* __restrict__ Wout,
                                   const float* __restrict__ bout,
                                   float* __restrict__ out) {
  const int lane = threadIdx.x & 31;
  const int wv   = threadIdx.x >> 5;
  const int l16  = lane & 15;
  const int lhi  = lane >> 4;
  const int m0   = blockIdx.x * 128 + wv * 16;

  v2f a[16];
#pragma unroll
  for (int kk = 0; kk < 16; ++kk)
    a[kk] = *(const v2f*)(outc + (size_t)(m0 + l16) * KDIM + kk * 4 + lhi * 2);

#pragma unroll 1
  for (int ct = 0; ct < HIDDEN / 16; ++ct) {            // 64 column tiles
    const int h = ct * 16 + l16;
    v8f C = {};
#pragma unroll
    for (int kk = 0; kk < 16; ++kk) {
      v2f bf = *(const v2f*)(Wout + (size_t)h * KDIM + kk * 4 + lhi * 2);
      C = wmma4(a[kk], bf, C);
    }
    const float bb = bout[h];
#pragma unroll
    for (int r = 0; r < 8; ++r)
      out[(size_t)(m0 + r + lhi * 8) * HIDDEN + h] = C[r] + bb;
  }
}

// ---------------------------------------------------------------------------
extern "C" void kernel_launch(void* const* d_in, const int* in_sizes, int n_in,
                              void* d_out, int out_size, void* d_ws, size_t ws_size,
                              hipStream_t stream) {
  const float* query = (const float*)d_in[0];
  const float* key   = (const float*)d_in[1];
  const float* value = (const float*)d_in[2];
  const float* Wq    = (const float*)d_in[3];
  const float* Wk    = (const float*)d_in[4];
  const float* Wv    = (const float*)d_in[5];
  const float* qm    = (const float*)d_in[6];
  const float* km    = (const float*)d_in[7];
  const float* vm    = (const float*)d_in[8];
  const float* Wout  = (const float*)d_in[9];
  const float* bout  = (const float*)d_in[10];

  float* out  = (float*)d_out;                       // (B,SQ,HIDDEN)
  float* attn = out + (size_t)NROWS * HIDDEN;        // (B,SQ,SK)

  // workspace: q_c | k_c | vT | out_c  -> 4 x 4MB = 16MB
  float* q_c  = (float*)d_ws;
  float* k_c  = q_c + (size_t)NROWS * KDIM;
  float* vT   = k_c + (size_t)NROWS * KDIM;
  float* outc = vT  + (size_t)NROWS * KDIM;

  eig_proj_kernel<<<dim3(NROWS / 128, 3), 256, 0, stream>>>(
      query, key, value, Wq, Wk, Wv, qm, km, vm, q_c, k_c, vT);

  eig_scores_kernel<<<dim3(SEQ / 128, SEQ / 128, BATCH), 256, 0, stream>>>(
      q_c, k_c, attn);

  eig_softmax_kernel<<<dim3(NROWS), 256, 0, stream>>>(attn);

  eig_outc_kernel<<<dim3(SEQ / 128, BATCH), 256, 0, stream>>>(attn, vT, outc);

  eig_outproj_kernel<<<dim3(NROWS / 128), 256, 0, stream>>>(outc, Wout, bout, out);
}